// IS2RE_GCN_37434934952476
// MI455X (gfx1250) — compile-verified
//
#include <hip/hip_runtime.h>
#include <hip/hip_bf16.h>

#define NN 50000
#define EE 800000
#define GG 512
#define HH 256
#define TT 32
#define LL 4
#define BN_EPS 1e-5f

typedef __attribute__((ext_vector_type(2))) float v2f;
typedef __attribute__((ext_vector_type(8))) float v8f;

// ---------------------------------------------------------------- utilities
__global__ void zero_f32(float* __restrict__ p, int n) {
    int i = blockIdx.x * blockDim.x + threadIdx.x;
    if (i < n) p[i] = 0.0f;
}

__global__ void edge_deg(const int* __restrict__ ei, float* __restrict__ deg) {
    int e = blockIdx.x * blockDim.x + threadIdx.x;
    if (e < EE) atomicAdd(&deg[ei[EE + e]], 1.0f);
}

__global__ void make_dinv(const float* __restrict__ deg, float* __restrict__ dinv) {
    int i = blockIdx.x * blockDim.x + threadIdx.x;
    if (i < NN) dinv[i] = rsqrtf(deg[i] + 1.0f);
}

// ------------------------------------------------------- fp32 WMMA GEMM core
// C[M,N] = A[M,K] @ B[K,N]; row-major, lda==K, ldb==N, ldc==N.
// Requires: K % 32 == 0, N % 64 == 0. Block tile 128(M) x 64(N), Kc = 32.
// 8 waves; wave w owns rows w*16..w*16+15 and all 64 cols (4 accumulators).
// LDS holds K-pair-contiguous float2 so every WMMA operand is one aligned
// 8-byte DS load straight into an even-aligned VGPR pair.
__global__ __launch_bounds__(256) void wmma_gemm_f32(
        const float* __restrict__ A, const float* __restrict__ B,
        float* __restrict__ C, int M, int N, int K) {
    __shared__ float2 As2[128][17];   // [row][kpair]; stride 34 floats (banks ok)
    __shared__ float2 Bs2[16][80];    // [kpair][n];  stride 160 floats (+32 banks)

    const int lane = threadIdx.x & 31;
    const int wid  = threadIdx.x >> 5;      // 0..7 -> M strip
    const int lm   = lane & 15;
    const int hi   = lane >> 4;             // 0: K+0/1, 1: K+2/3
    const int rowBase = blockIdx.y * 128;
    const int colBase = blockIdx.x * 64;

    v8f acc0 = {}, acc1 = {}, acc2 = {}, acc3 = {};

    for (int kb = 0; kb < K; kb += 32) {
        // stage A: 128 rows x 16 k-pairs (clamp rows for M tail; stores guarded)
        {
            const int pp = threadIdx.x & 15;
            for (int r = threadIdx.x >> 4; r < 128; r += 16) {
                int row = rowBase + r;
                if (row >= M) row = M - 1;
                As2[r][pp] =
                    *reinterpret_cast<const float2*>(A + (size_t)row * K + kb + 2 * pp);
            }
        }
        // stage B: 16 k-pairs x 64 cols
        {
            const int n = threadIdx.x & 63;
            for (int p = threadIdx.x >> 6; p < 16; p += 4) {
                float b0 = B[(size_t)(kb + 2 * p) * N + colBase + n];
                float b1 = B[(size_t)(kb + 2 * p + 1) * N + colBase + n];
                Bs2[p][n] = float2{b0, b1};
            }
        }
        __syncthreads();

#pragma unroll
        for (int kk = 0; kk < 32; kk += 4) {
            const int p = (kk >> 1) + hi;           // 2p == kk + kh
            float2 af  = As2[wid * 16 + lm][p];
            float2 bf0 = Bs2[p][lm];
            float2 bf1 = Bs2[p][16 + lm];
            float2 bf2 = Bs2[p][32 + lm];
            float2 bf3 = Bs2[p][48 + lm];
            v2f a, b;
            a.x = af.x;  a.y = af.y;
            b.x = bf0.x; b.y = bf0.y;
            acc0 = __builtin_amdgcn_wmma_f32_16x16x4_f32(
                false, a, false, b, (short)0, acc0, false, false);
            b.x = bf1.x; b.y = bf1.y;
            acc1 = __builtin_amdgcn_wmma_f32_16x16x4_f32(
                false, a, false, b, (short)0, acc1, false, false);
            b.x = bf2.x; b.y = bf2.y;
            acc2 = __builtin_amdgcn_wmma_f32_16x16x4_f32(
                false, a, false, b, (short)0, acc2, false, false);
            b.x = bf3.x; b.y = bf3.y;
            acc3 = __builtin_amdgcn_wmma_f32_16x16x4_f32(
                false, a, false, b, (short)0, acc3, false, false);
        }
        __syncthreads();
    }

    // C/D layout: VGPR r -> M = r (lanes 0-15) or r+8 (lanes 16-31), N = lm
    const int rb = rowBase + wid * 16 + (hi << 3);
#pragma unroll
    for (int r = 0; r < 8; ++r) {
        int row = rb + r;
        if (row < M) {
            float* cp = C + (size_t)row * N + colBase + lm;
            cp[0]  = acc0[r];
            cp[16] = acc1[r];
            cp[32] = acc2[r];
            cp[48] = acc3[r];
        }
    }
}

// ------------------------------------------------------------- GCN pipeline
// h[i,:] = P1[x[i],:] + P2[tags[i],:] + bp   (factored embedding projection)
__global__ void embed_h4(const int* __restrict__ x, const int* __restrict__ tags,
                         const float* __restrict__ P1, const float* __restrict__ P2,
                         const float* __restrict__ bp, float* __restrict__ h) {
    int node = blockIdx.x * 4 + (threadIdx.x >> 6);
    int c = (threadIdx.x & 63) * 4;
    float4 p1 = *(const float4*)(P1 + (size_t)x[node] * HH + c);
    float4 p2 = *(const float4*)(P2 + (size_t)tags[node] * HH + c);
    float4 bb = *(const float4*)(bp + c);
    float4 r{p1.x + p2.x + bb.x, p1.y + p2.y + bb.y,
             p1.z + p2.z + bb.z, p1.w + p2.w + bb.w};
    *(float4*)(h + (size_t)node * HH + c) = r;
}

// agg = hw * dinv^2   (self-loop term doubles as initialization)
__global__ void agg_init4(const float* __restrict__ hw, const float* __restrict__ dinv,
                          float* __restrict__ agg) {
    int node = blockIdx.x * 4 + (threadIdx.x >> 6);
    int c = (threadIdx.x & 63) * 4;
    float d = dinv[node];
    float s = d * d;
    float4 v = *(const float4*)(hw + (size_t)node * HH + c);
    float4 r{v.x * s, v.y * s, v.z * s, v.w * s};
    *(float4*)(agg + (size_t)node * HH + c) = r;
}

// agg[dst,:] += hw[src,:] * dinv[src]*dinv[dst]   (4 edges per block)
__global__ void scatter_edges4(const int* __restrict__ ei, const float* __restrict__ dinv,
                               const float* __restrict__ hw, float* __restrict__ agg) {
    int e = blockIdx.x * 4 + (threadIdx.x >> 6);
    int c = (threadIdx.x & 63) * 4;
    int s = ei[e], d = ei[EE + e];
    float norm = dinv[s] * dinv[d];
    float4 v = *(const float4*)(hw + (size_t)s * HH + c);
    float* ap = agg + (size_t)d * HH + c;
    atomicAdd(ap + 0, v.x * norm);
    atomicAdd(ap + 1, v.y * norm);
    atomicAdd(ap + 2, v.z * norm);
    atomicAdd(ap + 3, v.w * norm);
}

// per-column sum / sum-of-squares partial reduction (4 cols per thread)
__global__ void col_stats4(const float* __restrict__ agg, float* __restrict__ colsum,
                           float* __restrict__ colsq) {
    int c = (threadIdx.x & 63) * 4;
    int r0 = blockIdx.x * 4 + (threadIdx.x >> 6);
    int stride = gridDim.x * 4;
    float sx = 0, sy = 0, sz = 0, sw = 0;
    float qx = 0, qy = 0, qz = 0, qw = 0;
    for (int r = r0; r < NN; r += stride) {
        float4 v = *(const float4*)(agg + (size_t)r * HH + c);
        sx += v.x; sy += v.y; sz += v.z; sw += v.w;
        qx += v.x * v.x; qy += v.y * v.y; qz += v.z * v.z; qw += v.w * v.w;
    }
    atomicAdd(&colsum[c + 0], sx); atomicAdd(&colsum[c + 1], sy);
    atomicAdd(&colsum[c + 2], sz); atomicAdd(&colsum[c + 3], sw);
    atomicAdd(&colsq[c + 0], qx);  atomicAdd(&colsq[c + 1], qy);
    atomicAdd(&colsq[c + 2], qz);  atomicAdd(&colsq[c + 3], qw);
}

// h = relu(batchnorm(agg)) + h  (gcn_b omitted: cancelled by mean subtraction)
__global__ void bn_apply4(const float* __restrict__ agg, const float* __restrict__ colsum,
                          const float* __restrict__ colsq, const float* __restrict__ g,
                          const float* __restrict__ b, float* __restrict__ h) {
    int node = blockIdx.x * 4 + (threadIdx.x >> 6);
    int c = (threadIdx.x & 63) * 4;
    float4 cs = *(const float4*)(colsum + c);
    float4 cq = *(const float4*)(colsq + c);
    float4 gg = *(const float4*)(g + c);
    float4 bb = *(const float4*)(b + c);
    const float inv = 1.0f / NN;
    float mux = cs.x * inv, muy = cs.y * inv, muz = cs.z * inv, muw = cs.w * inv;
    float rx = rsqrtf(fmaxf(cq.x * inv - mux * mux, 0.0f) + BN_EPS);
    float ry = rsqrtf(fmaxf(cq.y * inv - muy * muy, 0.0f) + BN_EPS);
    float rz = rsqrtf(fmaxf(cq.z * inv - muz * muz, 0.0f) + BN_EPS);
    float rw = rsqrtf(fmaxf(cq.w * inv - muw * muw, 0.0f) + BN_EPS);
    size_t idx = (size_t)node * HH + c;
    float4 av = *(const float4*)(agg + idx);
    float4 hv = *(const float4*)(h + idx);
    float4 r{fmaxf((av.x - mux) * rx * gg.x + bb.x, 0.0f) + hv.x,
             fmaxf((av.y - muy) * ry * gg.y + bb.y, 0.0f) + hv.y,
             fmaxf((av.z - muz) * rz * gg.z + bb.z, 0.0f) + hv.z,
             fmaxf((av.w - muw) * rw * gg.w + bb.w, 0.0f) + hv.w};
    *(float4*)(h + idx) = r;
}

// global mean pool scatter
__global__ void pool_scatter4(const float* __restrict__ h, const int* __restrict__ batch,
                              float* __restrict__ gsum, float* __restrict__ gcnt) {
    int node = blockIdx.x * 4 + (threadIdx.x >> 6);
    int c = (threadIdx.x & 63) * 4;
    int g = batch[node];
    float4 v = *(const float4*)(h + (size_t)node * HH + c);
    float* gp = gsum + (size_t)g * HH + c;
    atomicAdd(gp + 0, v.x);
    atomicAdd(gp + 1, v.y);
    atomicAdd(gp + 2, v.z);
    atomicAdd(gp + 3, v.w);
    if ((threadIdx.x & 63) == 0) atomicAdd(&gcnt[g], 1.0f);
}

__global__ void pool_div(const float* __restrict__ gsum, const float* __restrict__ gcnt,
                         float* __restrict__ pooled) {
    int g = blockIdx.x, c = threadIdx.x;
    float cnt = fmaxf(gcnt[g], 1.0f);
    pooled[(size_t)g * HH + c] = gsum[(size_t)g * HH + c] / cnt;
}

__global__ void bias_relu(float* __restrict__ hid, const float* __restrict__ b1, int n) {
    int i = blockIdx.x * blockDim.x + threadIdx.x;
    if (i < n) {
        int c = i & (HH / 2 - 1);
        hid[i] = fmaxf(hid[i] + b1[c], 0.0f);
    }
}

// out[g] = hid[g,:] . W2 + b2   (one 128-thread block per graph)
__global__ void mlp_out(const float* __restrict__ hid, const float* __restrict__ W2,
                        const float* __restrict__ b2, float* __restrict__ out) {
    __shared__ float red[HH / 2];
    int g = blockIdx.x, t = threadIdx.x;
    red[t] = hid[(size_t)g * (HH / 2) + t] * W2[t];
    __syncthreads();
    for (int s = (HH / 4); s > 0; s >>= 1) {
        if (t < s) red[t] += red[t + s];
        __syncthreads();
    }
    if (t == 0) out[g] = red[0] + b2[0];
}

// ------------------------------------------------------------------- driver
extern "C" void kernel_launch(void* const* d_in, const int* in_sizes, int n_in,
                              void* d_out, int out_size, void* d_ws, size_t ws_size,
                              hipStream_t stream) {
    (void)in_sizes; (void)n_in; (void)out_size; (void)ws_size;

    const int*   x        = (const int*)d_in[0];
    const int*   tags     = (const int*)d_in[1];
    const int*   ei       = (const int*)d_in[2];
    const int*   batch    = (const int*)d_in[3];
    const float* atom_emb = (const float*)d_in[4];
    const float* tag_emb  = (const float*)d_in[5];
    const float* Wp       = (const float*)d_in[6];
    const float* bp       = (const float*)d_in[7];
    const float* gcn_W    = (const float*)d_in[8];
    /* d_in[9] = gcn_b: exactly cancelled by BatchNorm mean subtraction */
    const float* bn_g     = (const float*)d_in[10];
    const float* bn_b     = (const float*)d_in[11];
    const float* W1       = (const float*)d_in[12];
    const float* b1       = (const float*)d_in[13];
    const float* W2       = (const float*)d_in[14];
    const float* b2       = (const float*)d_in[15];
    float*       out      = (float*)d_out;

    float* ws = (float*)d_ws;
    float* h      = ws;  ws += (size_t)NN * HH;
    float* hw     = ws;  ws += (size_t)NN * HH;
    float* agg    = ws;  ws += (size_t)NN * HH;
    float* deg    = ws;  ws += NN;
    float* dinv   = ws;  ws += NN;
    float* P1     = ws;  ws += 100 * HH;
    float* P2     = ws;  ws += 4 * HH;
    float* colsum = ws;  ws += HH;          // colsum & colsq contiguous
    float* colsq  = ws;  ws += HH;
    float* gsum   = ws;  ws += (size_t)GG * HH;   // gsum & gcnt contiguous
    float* gcnt   = ws;  ws += GG;
    float* pooled = ws;  ws += (size_t)GG * HH;
    float* hid    = ws;  ws += GG * (HH / 2);

    // degrees with self-loops -> dinv
    zero_f32<<<(NN + 255) / 256, 256, 0, stream>>>(deg, NN);
    edge_deg<<<(EE + 255) / 256, 256, 0, stream>>>(ei, deg);
    make_dinv<<<(NN + 255) / 256, 256, 0, stream>>>(deg, dinv);

    // factored input projection: P1 = atom_emb @ Wp[0:256], P2 = tag_emb @ Wp[256:288]
    wmma_gemm_f32<<<dim3(HH / 64, 1), 256, 0, stream>>>(atom_emb, Wp, P1, 100, HH, HH);
    wmma_gemm_f32<<<dim3(HH / 64, 1), 256, 0, stream>>>(
        tag_emb, Wp + (size_t)HH * HH, P2, 3, HH, TT);
    embed_h4<<<NN / 4, 256, 0, stream>>>(x, tags, P1, P2, bp, h);

    for (int l = 0; l < LL; ++l) {
        wmma_gemm_f32<<<dim3(HH / 64, (NN + 127) / 128), 256, 0, stream>>>(
            h, gcn_W + (size_t)l * HH * HH, hw, NN, HH, HH);
        agg_init4<<<NN / 4, 256, 0, stream>>>(hw, dinv, agg);
        scatter_edges4<<<EE / 4, 256, 0, stream>>>(ei, dinv, hw, agg);
        zero_f32<<<(2 * HH + 255) / 256, 256, 0, stream>>>(colsum, 2 * HH);
        col_stats4<<<512, 256, 0, stream>>>(agg, colsum, colsq);
        bn_apply4<<<NN / 4, 256, 0, stream>>>(agg, colsum, colsq,
                                              bn_g + (size_t)l * HH,
                                              bn_b + (size_t)l * HH, h);
    }

    // global mean pool + MLP head
    zero_f32<<<((GG * HH + GG) + 255) / 256, 256, 0, stream>>>(gsum, GG * HH + GG);
    pool_scatter4<<<NN / 4, 256, 0, stream>>>(h, batch, gsum, gcnt);
    pool_div<<<GG, HH, 0, stream>>>(gsum, gcnt, pooled);
    wmma_gemm_f32<<<dim3((HH / 2) / 64, (GG + 127) / 128), 256, 0, stream>>>(
        pooled, W1, hid, GG, HH / 2, HH);
    bias_relu<<<(GG * (HH / 2) + 255) / 256, 256, 0, stream>>>(hid, b1, GG * (HH / 2));
    mlp_out<<<GG, HH / 2, 0, stream>>>(hid, W2, b2, out);
}